// PainnMessage_38001870635382
// MI455X (gfx1250) — compile-verified
//
#include <hip/hip_runtime.h>
#include <hip/hip_bf16.h>
#include <math.h>

#define NODE_DIM   128
#define NUM_IRREPS 224
#define SPH_DIM    480
#define HIDDEN_DIM 576   // NODE_DIM + 2*NUM_IRREPS
#define NUM_BASIS  20
#define N_NODES    10000
#define N_EDGES    160000

typedef __attribute__((ext_vector_type(2))) float v2f;
typedef __attribute__((ext_vector_type(8))) float v8f;

// FP32 WMMA: D(16x16) = A(16x4) * B(4x16) + C  -> v_wmma_f32_16x16x4_f32
__device__ __forceinline__ v8f wmma16x16x4(v2f a, v2f b, v8f c) {
  return __builtin_amdgcn_wmma_f32_16x16x4_f32(false, a, false, b, (short)0, c,
                                               false, false);
}

__device__ __forceinline__ float silu_f(float v) {
  return v / (1.0f + __expf(-v));
}

// ---------------------------------------------------------------------------
// Kernel 0: seed outputs with x_scalar / x_spherical (scatter-add baseline)
// ---------------------------------------------------------------------------
__global__ void painn_init_out(const float* __restrict__ xs,
                               const float* __restrict__ xsph,
                               float* __restrict__ out) {
  const int n1 = N_NODES * NODE_DIM;   // 1,280,000
  const int n2 = N_NODES * SPH_DIM;    // 4,800,000
  const int stride = gridDim.x * blockDim.x;
  int i = blockIdx.x * blockDim.x + threadIdx.x;
  const float4* s1 = (const float4*)xs;
  const float4* s2 = (const float4*)xsph;
  float4* o1 = (float4*)out;
  float4* o2 = (float4*)(out + n1);
  for (int idx = i; idx < n1 / 4; idx += stride) o1[idx] = s1[idx];
  for (int idx = i; idx < n2 / 4; idx += stride) o2[idx] = s2[idx];
}

// ---------------------------------------------------------------------------
// Kernel 1: H = silu(x_scalar @ W1 + b1)      [N_NODES x 128]
// grid.x = N_NODES/16 row tiles; 8 waves, wave w owns column tile w (128 = 8*16)
// ---------------------------------------------------------------------------
__global__ void __launch_bounds__(256) painn_gemm1(const float* __restrict__ xs,
                                                   const float* __restrict__ W1,
                                                   const float* __restrict__ b1,
                                                   float* __restrict__ H) {
  const int wave = threadIdx.x >> 5;
  const int lane = threadIdx.x & 31;
  const int half = lane >> 4;     // 0: K={0,1}  1: K={2,3}
  const int lid  = lane & 15;     // M (for A) / N (for B,C,D)
  const int m0 = blockIdx.x << 4;
  const int n0 = wave << 4;

  v8f acc = {};
  const float* ap = xs + (m0 + lid) * NODE_DIM + 2 * half;
  const float* bp = W1 + (2 * half) * NODE_DIM + n0 + lid;
  for (int k0 = 0; k0 < NODE_DIM; k0 += 4) {
    v2f a, b;
    a.x = ap[k0];
    a.y = ap[k0 + 1];
    b.x = bp[k0 * NODE_DIM];
    b.y = bp[(k0 + 1) * NODE_DIM];
    acc = wmma16x16x4(a, b, acc);
  }
  const float bias = b1[n0 + lid];
#pragma unroll
  for (int i = 0; i < 8; ++i) {
    const int m = m0 + i + 8 * half;
    H[m * NODE_DIM + n0 + lid] = silu_f(acc[i] + bias);
  }
}

// ---------------------------------------------------------------------------
// Kernel 2: S = H @ W2 + b2                   [N_NODES x 576]
// grid.x = N_NODES/16 row tiles; 8 waves sweep the 36 column tiles
// ---------------------------------------------------------------------------
__global__ void __launch_bounds__(256) painn_gemm2(const float* __restrict__ H,
                                                   const float* __restrict__ W2,
                                                   const float* __restrict__ b2,
                                                   float* __restrict__ S) {
  const int wave = threadIdx.x >> 5;
  const int lane = threadIdx.x & 31;
  const int half = lane >> 4;
  const int lid  = lane & 15;
  const int m0 = blockIdx.x << 4;

  for (int t = wave; t < HIDDEN_DIM / 16; t += 8) {
    const int n0 = t << 4;
    v8f acc = {};
    const float* ap = H + (m0 + lid) * NODE_DIM + 2 * half;
    const float* bp = W2 + (2 * half) * HIDDEN_DIM + n0 + lid;
    for (int k0 = 0; k0 < NODE_DIM; k0 += 4) {
      v2f a, b;
      a.x = ap[k0];
      a.y = ap[k0 + 1];
      b.x = bp[k0 * HIDDEN_DIM];
      b.y = bp[(k0 + 1) * HIDDEN_DIM];
      acc = wmma16x16x4(a, b, acc);
    }
    const float bias = b2[n0 + lid];
#pragma unroll
    for (int i = 0; i < 8; ++i) {
      const int m = m0 + i + 8 * half;
      S[m * HIDDEN_DIM + n0 + lid] = acc[i] + bias;
    }
  }
}

// ---------------------------------------------------------------------------
// Kernel 3: fused edge kernel, 16 edges per workgroup.
//   filter_out(16x576) = (rbf_tile @ Wr + br) * fcut * S[dst]   (WMMA, K=20)
//   then gate-expand + scatter-add messages with global f32 atomics.
// ---------------------------------------------------------------------------
__global__ void __launch_bounds__(256) painn_edge(
    const float* __restrict__ S,     // [N_NODES x 576]
    const float* __restrict__ rbf,   // [N_EDGES x 20]
    const float* __restrict__ fcut,  // [N_EDGES]
    const float* __restrict__ rsh,   // [N_EDGES x 480]
    const float* __restrict__ xsph,  // [N_NODES x 480]
    const int* __restrict__ eidx,    // [2 x N_EDGES] (src row, dst row)
    const float* __restrict__ Wr,    // [20 x 576]
    const float* __restrict__ br,    // [576]
    float* __restrict__ out_scalar,  // [N_NODES x 128]
    float* __restrict__ out_sph) {   // [N_NODES x 480]
  __shared__ float s_fo[16 * HIDDEN_DIM];   // 36 KB filter_out tile
  __shared__ float s_rbf[16 * NUM_BASIS];
  __shared__ int s_src[16];
  __shared__ int s_dst[16];
  __shared__ float s_fcut[16];

  const int tid = threadIdx.x;
  const int e0 = blockIdx.x << 4;

  if (tid < 16) {
    s_src[tid] = eidx[e0 + tid];
    s_dst[tid] = eidx[N_EDGES + e0 + tid];
    s_fcut[tid] = fcut[e0 + tid];
  }
  for (int i = tid; i < 16 * NUM_BASIS; i += 256)
    s_rbf[i] = rbf[e0 * NUM_BASIS + i];
  __syncthreads();

  const int wave = tid >> 5;
  const int lane = tid & 31;
  const int half = lane >> 4;
  const int lid  = lane & 15;

  // --- filter GEMM: M=16 edges, N=576 (36 tiles over 8 waves), K=20 ---
  for (int t = wave; t < HIDDEN_DIM / 16; t += 8) {
    const int n0 = t << 4;
    v8f acc = {};
    const float* ap = s_rbf + lid * NUM_BASIS + 2 * half;
    const float* bp = Wr + (2 * half) * HIDDEN_DIM + n0 + lid;
#pragma unroll
    for (int k0 = 0; k0 < NUM_BASIS; k0 += 4) {   // 5 WMMA steps
      v2f a, b;
      a.x = ap[k0];
      a.y = ap[k0 + 1];
      b.x = bp[k0 * HIDDEN_DIM];
      b.y = bp[(k0 + 1) * HIDDEN_DIM];
      acc = wmma16x16x4(a, b, acc);
    }
    const float bias = br[n0 + lid];
#pragma unroll
    for (int i = 0; i < 8; ++i) {
      const int m = i + 8 * half;  // edge within tile
      const float fw = (acc[i] + bias) * s_fcut[m];
      const float fo = S[(size_t)s_dst[m] * HIDDEN_DIM + n0 + lid] * fw;
      s_fo[m * HIDDEN_DIM + n0 + lid] = fo;
    }
  }
  __syncthreads();

  // --- message_scalar: columns [448,576) scatter-add into out_scalar[src] ---
  for (int idx = tid; idx < 16 * NODE_DIM; idx += 256) {
    const int e = idx >> 7;
    const int j = idx & (NODE_DIM - 1);
    atomicAdd(&out_scalar[(size_t)s_src[e] * NODE_DIM + j],
              s_fo[e * HIDDEN_DIM + 2 * NUM_IRREPS + j]);
  }

  // --- message_spherical: gate expansion (128x1 + 64x3 + 32x5) + scatter ---
  for (int e = 0; e < 16; ++e) {
    const size_t dbase = (size_t)s_dst[e] * SPH_DIM;
    const size_t sbase = (size_t)s_src[e] * SPH_DIM;
    const size_t rbase = (size_t)(e0 + e) * SPH_DIM;
    const float* fo = s_fo + e * HIDDEN_DIM;
    for (int j = tid; j < SPH_DIM; j += 256) {
      int g;
      if (j < 128)      g = j;
      else if (j < 320) g = 128 + (j - 128) / 3;
      else              g = 192 + (j - 320) / 5;
      const float gs = fo[g];
      const float ge = fo[NUM_IRREPS + g];
      const float msg = xsph[dbase + j] * gs + rsh[rbase + j] * ge;
      atomicAdd(&out_sph[sbase + j], msg);
    }
  }
}

// ---------------------------------------------------------------------------
extern "C" void kernel_launch(void* const* d_in, const int* in_sizes, int n_in,
                              void* d_out, int out_size, void* d_ws,
                              size_t ws_size, hipStream_t stream) {
  const float* xs   = (const float*)d_in[0];
  const float* xsph = (const float*)d_in[1];
  const float* rbf  = (const float*)d_in[2];
  const float* fcut = (const float*)d_in[3];
  const float* rsh  = (const float*)d_in[4];
  const int*   eidx = (const int*)d_in[5];
  const float* W1   = (const float*)d_in[6];
  const float* b1   = (const float*)d_in[7];
  const float* W2   = (const float*)d_in[8];
  const float* b2   = (const float*)d_in[9];
  const float* Wr   = (const float*)d_in[10];
  const float* br   = (const float*)d_in[11];

  float* H = (float*)d_ws;                          // [N_NODES x 128]  5.1 MB
  float* S = H + (size_t)N_NODES * NODE_DIM;        // [N_NODES x 576] 23.0 MB
  float* out_scalar = (float*)d_out;
  float* out_sph = out_scalar + (size_t)N_NODES * NODE_DIM;

  painn_init_out<<<1024, 256, 0, stream>>>(xs, xsph, (float*)d_out);
  painn_gemm1<<<N_NODES / 16, 256, 0, stream>>>(xs, W1, b1, H);
  painn_gemm2<<<N_NODES / 16, 256, 0, stream>>>(H, W2, b2, S);
  painn_edge<<<N_EDGES / 16, 256, 0, stream>>>(S, rbf, fcut, rsh, xsph, eidx,
                                               Wr, br, out_scalar, out_sph);
}